// Entmax_58342835748885
// MI455X (gfx1250) — compile-verified
//
#include <hip/hip_runtime.h>
#include <stdint.h>

// entmax-1.5 along last dim (D=1024) for f32 input of shape [..., 1024].
// One wave32 per row, 32 elements per lane kept in VGPRs.
// Tile staging: Tensor Data Mover (tensor_load_to_lds) global -> LDS,
// then ds_load_b128 into registers. tau found by monotone Newton on
// f(tau) = sum(relu(u - tau)^2) - 1  (convex, decreasing, start tau=-1).

typedef unsigned int u32x4 __attribute__((ext_vector_type(4)));
typedef int          i32x4 __attribute__((ext_vector_type(4)));
typedef int          i32x8 __attribute__((ext_vector_type(8)));

#define ROW_D        1024
#define ROWS_PER_WG  8          // one row per wave, 8 waves = 256 threads
#define NEWTON_ITERS 20

__global__ __launch_bounds__(256)
void entmax15_tdm_kernel(const float* __restrict__ x,
                         float* __restrict__ y,
                         int nrows) {
    __shared__ float tile[ROWS_PER_WG * ROW_D];     // 32 KB of 320 KB LDS

    const int wave = threadIdx.x >> 5;
    const int lane = threadIdx.x & 31;
    const long long row0 = (long long)blockIdx.x * ROWS_PER_WG;

    // ---------------- TDM: DMA an 8x1024 f32 tile into LDS ----------------
    if (threadIdx.x == 0) {
        uint64_t gaddr = (uint64_t)(uintptr_t)(const void*)(x + row0 * ROW_D);
        uint32_t ldsa  = (uint32_t)(uintptr_t)(void*)tile;   // low 32 bits = LDS byte addr

        // D# group 0 (128b): count=1 | lds_addr | global_addr[56:0] | type=2
        u32x4 g0;
        g0[0] = 1u;                                        // count = 1 (valid)
        g0[1] = ldsa;                                      // lds_addr
        g0[2] = (uint32_t)gaddr;                           // global_addr[31:0]
        g0[3] = (uint32_t)((gaddr >> 32) & 0x01FFFFFFu)    // global_addr[56:32]
              | (2u << 30);                                // type = 2 ("image")

        // D# group 1 (256b), packed as 4x u64 then split to dwords:
        // q0: workgroup_mask=0 | data_size=2 (4B) | flags=0 | tensor_dim0[15:0]<<48
        uint64_t q0 = (2ull << 16) | ((uint64_t)(ROW_D & 0xFFFF) << 48);
        // q1: tensor_dim0[31:16] | tensor_dim1<<16 | tile_dim0<<48
        uint64_t q1 = ((uint64_t)(ROW_D >> 16))
                    | ((uint64_t)ROWS_PER_WG << 16)
                    | ((uint64_t)ROW_D << 48);
        // q2: tile_dim1 | tile_dim2=0 | tensor_dim0_stride[31:0]<<32
        uint64_t q2 = (uint64_t)ROWS_PER_WG | ((uint64_t)ROW_D << 32);
        // q3: tensor_dim0_stride[47:32]=0 | tensor_dim1_stride=0 (2D tile)
        uint64_t q3 = 0;

        i32x8 g1;
        g1[0] = (int)(uint32_t)q0;  g1[1] = (int)(uint32_t)(q0 >> 32);
        g1[2] = (int)(uint32_t)q1;  g1[3] = (int)(uint32_t)(q1 >> 32);
        g1[4] = (int)(uint32_t)q2;  g1[5] = (int)(uint32_t)(q2 >> 32);
        g1[6] = (int)(uint32_t)q3;  g1[7] = (int)(uint32_t)(q3 >> 32);

        i32x4 z4; z4[0] = 0; z4[1] = 0; z4[2] = 0; z4[3] = 0;  // groups 2/3 unused (2D)
        i32x8 z8;
        z8[0] = 0; z8[1] = 0; z8[2] = 0; z8[3] = 0;
        z8[4] = 0; z8[5] = 0; z8[6] = 0; z8[7] = 0;

        // 6-arg form (clang-23 / therock-10.0): (g0, g1, g2, g3, extra, cpol)
        __builtin_amdgcn_tensor_load_to_lds(g0, g1, z4, z4, z8, /*cpol=*/0);
    }
    __builtin_amdgcn_s_wait_tensorcnt(0);   // issuing wave waits; others: no-op
    __syncthreads();                        // tile visible to all waves

    const bool live = (row0 + wave) < (long long)nrows;

    // -------- unpack row from LDS: lane holds elements 4*lane+128k+{0..3} --------
    const float4* lp = (const float4*)(tile + wave * ROW_D);
    float u[32];
    float m = -3.402823466e+38f;
#pragma unroll
    for (int k = 0; k < 8; ++k) {
        float4 a = lp[lane + 32 * k];
        u[4*k+0] = a.x * 0.5f;  u[4*k+1] = a.y * 0.5f;
        u[4*k+2] = a.z * 0.5f;  u[4*k+3] = a.w * 0.5f;
        m = fmaxf(m, fmaxf(fmaxf(u[4*k+0], u[4*k+1]), fmaxf(u[4*k+2], u[4*k+3])));
    }
    // wave-wide max (wave32 butterfly)
#pragma unroll
    for (int off = 16; off > 0; off >>= 1)
        m = fmaxf(m, __shfl_xor(m, off, 32));
#pragma unroll
    for (int i = 0; i < 32; ++i) u[i] -= m;

    // -------- monotone Newton for tau in [-1, 0]:  f(tau)=sum(relu(u-tau)^2)-1 --------
    float tau = -1.0f;
#pragma unroll 1
    for (int it = 0; it < NEWTON_ITERS; ++it) {
        float s1 = 0.0f, s2 = 0.0f;
#pragma unroll
        for (int i = 0; i < 32; ++i) {
            float t = fmaxf(u[i] - tau, 0.0f);
            s1 += t;
            s2 = fmaf(t, t, s2);
        }
#pragma unroll
        for (int off = 16; off > 0; off >>= 1) {
            s1 += __shfl_xor(s1, off, 32);
            s2 += __shfl_xor(s2, off, 32);
        }
        // tau_{k+1} = tau - f/f' = tau + (s2-1)/(2*s1); f>=0 and convex => no overshoot
        tau += (s2 - 1.0f) / (2.0f * fmaxf(s1, 1e-20f));
    }

    // -------- emit y = relu(u - tau)^2, coalesced b128 stores --------
    if (live) {
        float4* op = (float4*)(y + (row0 + wave) * ROW_D);
#pragma unroll
        for (int k = 0; k < 8; ++k) {
            float4 r;
            float t0 = fmaxf(u[4*k+0] - tau, 0.0f);
            float t1 = fmaxf(u[4*k+1] - tau, 0.0f);
            float t2 = fmaxf(u[4*k+2] - tau, 0.0f);
            float t3 = fmaxf(u[4*k+3] - tau, 0.0f);
            r.x = t0 * t0;  r.y = t1 * t1;  r.z = t2 * t2;  r.w = t3 * t3;
            op[lane + 32 * k] = r;
        }
    }
}

extern "C" void kernel_launch(void* const* d_in, const int* in_sizes, int n_in,
                              void* d_out, int out_size, void* d_ws, size_t ws_size,
                              hipStream_t stream) {
    const float* x = (const float*)d_in[0];
    float* y = (float*)d_out;
    const int n     = in_sizes[0];          // 8*16*1024*1024 = 134217728
    const int nrows = n / ROW_D;            // 131072
    const int blocks = (nrows + ROWS_PER_WG - 1) / ROWS_PER_WG;
    entmax15_tdm_kernel<<<blocks, 256, 0, stream>>>(x, y, nrows);
}